// Kmeans_Batch_712964571137
// MI455X (gfx1250) — compile-verified
//
#include <hip/hip_runtime.h>

// ---------------------------------------------------------------------------
// VQ loss for MI455X (gfx1250):
//   loss = (1/(N*D)) * sum_n ( ||x_n||^2 + min_k( ||w_k||^2 - 2 * x_n . w_k ) )
// Core: f32 GEMM X[32768,512] x W^T[512,1024] on V_WMMA_F32_16X16X4_F32.
// W tiles (and the X tile) are fetched by the Tensor Data Mover
// (TENSOR_LOAD_TO_LDS) with double buffering + s_wait_tensorcnt, using TDM's
// LDS pad feature (2 dwords per 256 dwords) for a 516-float row stride.
// NOTE: descriptor lds_addr values are layout CONSTANTS (dynamic LDS starts
// at offset 0) -- no ptrtoint on the LDS pointers, so address-space inference
// keeps all fragment reads on the DS path (ds_load_2addr_b64), not FLAT.
// ---------------------------------------------------------------------------

typedef __attribute__((ext_vector_type(2))) float v2f;
typedef __attribute__((ext_vector_type(8))) float v8f;
typedef __attribute__((ext_vector_type(4))) unsigned int u32x4;
typedef __attribute__((ext_vector_type(8))) int i32x8;
typedef __attribute__((ext_vector_type(4))) int i32x4;

#define EMB_D      512      // embedding dim (contraction)
#define KCODES     1024     // codebook entries
#define NROWS      32768    // 64*512 flattened rows
#define ROWS_BLK   64       // rows per workgroup (4 waves x 16)
#define THREADS    128
#define NTILES     (KCODES / 16)
// LDS row stride in floats: 512 data + TDM pad (2 dwords after every 256
// dwords -> pads land at d=256 and row end). Rows 16B aligned; 4-bank row
// stride => conflict-free b64 fragment reads across a 16-lane half.
#define LSTR       516

// LDS layout (float offsets from dynamic-LDS base, which is offset 0):
#define XOFF_F     0                    // X tile: 64 * LSTR
#define W0OFF_F    (64 * LSTR)          // W ping: 16 * LSTR
#define W1OFF_F    (80 * LSTR)          // W pong: 16 * LSTR
#define WNOFF_F    (96 * LSTR)          // wnorm: 1024
#define REDOFF_F   (WNOFF_F + KCODES)   // 4 wave partials

// ---------------------------- wnorm: ||w_k||^2 ------------------------------
__global__ void vq_wnorm_kernel(const float* __restrict__ W, float* __restrict__ wn) {
    int k = blockIdx.x * blockDim.x + threadIdx.x;   // 1024 threads total
    const float4* row = reinterpret_cast<const float4*>(W + (size_t)k * EMB_D);
    float s = 0.f;
#pragma unroll 8
    for (int i = 0; i < EMB_D / 4; ++i) {
        float4 v = row[i];
        s += v.x * v.x + v.y * v.y + v.z * v.z + v.w * v.w;
    }
    wn[k] = s;
}

// ---------------- TDM: 2D f32 tile (rows x 512) global -> LDS ---------------
// D# per CDNA5 ISA §8.3/8.4. data_size=4B, pad_enable=1, pad_interval=7
// (256 dwords), pad_amount=1 (2 dwords) -> LDS row stride 516 floats.
__device__ __forceinline__ void tdm_load_tile_f32(const float* gsrc,
                                                  unsigned ldsByteOff,
                                                  int tileRows) {
    unsigned long long ga = (unsigned long long)(uintptr_t)gsrc;
    u32x4 g0;
    g0[0] = 1u;                                            // count=1 (valid)
    g0[1] = ldsByteOff;                                    // lds_addr (const)
    g0[2] = (unsigned)(ga & 0xFFFFFFFFu);                  // global_addr lo
    g0[3] = (unsigned)((ga >> 32) & 0x01FFFFFFu)           // global_addr hi
          | 0x80000000u;                                   // type=2 (image)

    i32x8 g1;
    g1[0] = (int)((2u << 16)        // data_size = 4 bytes
                | (1u << 20)        // pad_enable
                | (7u << 22)        // pad_interval = 256 dwords
                | (1u << 25));      // pad_amount   = 2 dwords
    g1[1] = (int)(512u << 16);      // tensor_dim0 = 512 (bits[79:48] lo half)
    g1[2] = (int)((unsigned)tileRows << 16);  // tensor_dim1 lo16
    g1[3] = (int)(512u << 16);      // tile_dim0 = 512 (bits[127:112])
    g1[4] = tileRows;               // tile_dim1 (tile_dim2 = 0)
    g1[5] = 512;                    // tensor_dim0_stride = 512 elements
    g1[6] = 0;
    g1[7] = 0;

    i32x4 z4 = {0, 0, 0, 0};                    // groups 2/3 unused (<=2D)
    i32x8 z8 = {0, 0, 0, 0, 0, 0, 0, 0};        // extra group (unused)
    __builtin_amdgcn_tensor_load_to_lds(g0, g1, z4, z4, z8, 0);
}

// ------------------- main fused GEMM + row-min kernel -----------------------
__global__ void vq_main_kernel(const float* __restrict__ X,
                               const float* __restrict__ W,
                               const float* __restrict__ wnorm,
                               float* __restrict__ blockPartials) {
    extern __shared__ float smem[];

    const int t    = threadIdx.x;
    const int lane = t & 31;
    const int wave = t >> 5;
    const int l16  = lane & 15;
    const int hi   = lane >> 4;                // which 16-lane half
    const int rowBase = blockIdx.x * ROWS_BLK;

    // Kick off DMAs: X tile + first two W tiles (wave 0 issues; in-order).
    // lds_addr fields are layout constants -- LDS pointers never escape.
    if (wave == 0) {
        tdm_load_tile_f32(X + (size_t)rowBase * EMB_D, XOFF_F * 4u, ROWS_BLK);
        tdm_load_tile_f32(W,                           W0OFF_F * 4u, 16);
        tdm_load_tile_f32(W + 16 * EMB_D,              W1OFF_F * 4u, 16);
    }

    // codebook norms -> LDS (overlaps the DMAs; uniform loop, no divergence)
#pragma unroll
    for (int i = 0; i < KCODES / THREADS; ++i)
        smem[WNOFF_F + t + i * THREADS] = wnorm[t + i * THREADS];

    __builtin_amdgcn_s_wait_tensorcnt(1);   // X + W tile 0 landed (wave 0)
    __syncthreads();

    // Per-lane partial ||x_row||^2 over this lane's A-fragment element set
    // (d % 4 in {2hi, 2hi+1}); partner half-lane holds the complement.
    const float* aBase = smem + XOFF_F + (wave * 16 + l16) * LSTR + 2 * hi;
    float xnp = 0.f;
#pragma unroll 16
    for (int s = 0; s < EMB_D / 4; ++s) {
        const int off = 4 * s + ((s >= 64) ? 2 : 0);   // TDM pad at d=256
        v2f a = *reinterpret_cast<const v2f*>(aBase + off);
        xnp += a.x * a.x + a.y * a.y;
    }
    float xnorm = xnp + __shfl_xor(xnp, 16, 32);

    float minv[8];
#pragma unroll
    for (int v = 0; v < 8; ++v) minv[v] = 3.402823466e38f;

    for (int nt = 0; nt < NTILES; ++nt) {
        // ping/pong select by integer offset (keeps one clean LDS ptr chain)
        const int wOff = W0OFF_F + (nt & 1) * (16 * LSTR);
        const float* bBase = smem + wOff + l16 * LSTR + 2 * hi;

        // 4 independent accumulator chains over 128 K=4 steps (matrix-pipe
        // ILP: 202KB LDS -> 1 WG/WGP -> 1 wave/SIMD, so ILP is intra-wave).
        v8f acc0 = {}, acc1 = {}, acc2 = {}, acc3 = {};
#pragma unroll
        for (int s = 0; s < EMB_D / 16; ++s) {
            const int o = 16 * s + ((s >= 16) ? 2 : 0);  // TDM pad at d=256
            v2f a0 = *reinterpret_cast<const v2f*>(aBase + o + 0);
            v2f b0 = *reinterpret_cast<const v2f*>(bBase + o + 0);
            acc0 = __builtin_amdgcn_wmma_f32_16x16x4_f32(false, a0, false, b0,
                                                         (short)0, acc0, false, false);
            v2f a1 = *reinterpret_cast<const v2f*>(aBase + o + 4);
            v2f b1 = *reinterpret_cast<const v2f*>(bBase + o + 4);
            acc1 = __builtin_amdgcn_wmma_f32_16x16x4_f32(false, a1, false, b1,
                                                         (short)0, acc1, false, false);
            v2f a2 = *reinterpret_cast<const v2f*>(aBase + o + 8);
            v2f b2 = *reinterpret_cast<const v2f*>(bBase + o + 8);
            acc2 = __builtin_amdgcn_wmma_f32_16x16x4_f32(false, a2, false, b2,
                                                         (short)0, acc2, false, false);
            v2f a3 = *reinterpret_cast<const v2f*>(aBase + o + 12);
            v2f b3 = *reinterpret_cast<const v2f*>(bBase + o + 12);
            acc3 = __builtin_amdgcn_wmma_f32_16x16x4_f32(false, a3, false, b3,
                                                         (short)0, acc3, false, false);
        }

        // dist = ||w||^2 - 2*S ; wnorm depends only on this lane's column.
        float wn = smem[WNOFF_F + nt * 16 + l16];
#pragma unroll
        for (int v = 0; v < 8; ++v) {
            float s4 = (acc0[v] + acc1[v]) + (acc2[v] + acc3[v]);
            minv[v] = fminf(minv[v], __builtin_fmaf(-2.f, s4, wn));
        }

        // Pipeline: everyone done reading buf[nt&1] -> refill it 2 tiles ahead,
        // then guarantee buf[(nt+1)&1] (tile nt+1) has fully landed.
        __syncthreads();
        if (nt + 2 < NTILES) {
            if (wave == 0)
                tdm_load_tile_f32(W + (size_t)(nt + 2) * 16 * EMB_D,
                                  (unsigned)(W0OFF_F + (nt & 1) * (16 * LSTR)) * 4u,
                                  16);
            __builtin_amdgcn_s_wait_tensorcnt(1);   // tile nt+1 complete
        } else {
            __builtin_amdgcn_s_wait_tensorcnt(0);   // tail: drain everything
        }
        __syncthreads();
    }

    // min over the 16 columns held by this half (rows v + 8*hi)
#pragma unroll
    for (int v = 0; v < 8; ++v) {
        float m = minv[v];
        m = fminf(m, __shfl_xor(m, 1, 32));
        m = fminf(m, __shfl_xor(m, 2, 32));
        m = fminf(m, __shfl_xor(m, 4, 32));
        m = fminf(m, __shfl_xor(m, 8, 32));
        minv[v] = m;
    }
    float sm = 0.f;
#pragma unroll
    for (int v = 0; v < 8; ++v) sm += minv[v];

    // xnorm duplicated across halves (x0.5); sm duplicated across the 16
    // lanes of its half (x1/16). 32-lane sum -> exact per-wave total,
    // fixed order -> deterministic.
    float contrib = 0.5f * xnorm + sm * 0.0625f;
    contrib += __shfl_xor(contrib, 1, 32);
    contrib += __shfl_xor(contrib, 2, 32);
    contrib += __shfl_xor(contrib, 4, 32);
    contrib += __shfl_xor(contrib, 8, 32);
    contrib += __shfl_xor(contrib, 16, 32);

    if (lane == 0) smem[REDOFF_F + wave] = contrib;
    __syncthreads();
    if (t == 0)
        blockPartials[blockIdx.x] = (smem[REDOFF_F + 0] + smem[REDOFF_F + 1])
                                  + (smem[REDOFF_F + 2] + smem[REDOFF_F + 3]);
}

// -------------------- final deterministic reduction -------------------------
__global__ void vq_reduce_kernel(const float* __restrict__ partials,
                                 float* __restrict__ out) {
    __shared__ float s[256];
    int t = threadIdx.x;
    s[t] = partials[t] + partials[t + 256];   // 512 partials
    __syncthreads();
    for (int off = 128; off > 0; off >>= 1) {
        if (t < off) s[t] += s[t + off];
        __syncthreads();
    }
    if (t == 0)
        out[0] = s[0] * (1.0f / ((float)NROWS * (float)EMB_D));
}

// ---------------------------------------------------------------------------
extern "C" void kernel_launch(void* const* d_in, const int* in_sizes, int n_in,
                              void* d_out, int out_size, void* d_ws, size_t ws_size,
                              hipStream_t stream) {
    const float* X = (const float*)d_in[0];   // [32768, 512]
    const float* W = (const float*)d_in[1];   // [1024, 512]
    float* out = (float*)d_out;

    float* wn       = (float*)d_ws;           // 1024 floats
    float* partials = wn + KCODES;            // 512 floats

    hipLaunchKernelGGL(vq_wnorm_kernel, dim3(KCODES / 256), dim3(256), 0, stream, W, wn);

    // 64*516 (X) + 2*16*516 (W ping/pong) + 1024 (wnorm) + 8 (reduce)
    size_t ldsBytes = (size_t)(REDOFF_F + 8) * sizeof(float);
    hipLaunchKernelGGL(vq_main_kernel, dim3(NROWS / ROWS_BLK), dim3(THREADS),
                       ldsBytes, stream, X, W, wn, partials);

    hipLaunchKernelGGL(vq_reduce_kernel, dim3(1), dim3(256), 0, stream, partials, out);
}